// GKATNet_19670950215686
// MI455X (gfx1250) — compile-verified
//
#include <hip/hip_runtime.h>
#include <hip/hip_bf16.h>

typedef float v2f __attribute__((ext_vector_type(2)));
typedef float v8f __attribute__((ext_vector_type(8)));
typedef uint32_t u32x4 __attribute__((ext_vector_type(4)));
typedef int i32x4 __attribute__((ext_vector_type(4)));
typedef int i32x8 __attribute__((ext_vector_type(8)));

#define kN 2048
#define kE 65536
#define kCIN 128
#define kC 128
#define kH 8
#define NEG_SLOPE 0.2f

#define TM 128
#define TN 128
#define TK 32

// ---------------------------------------------------------------------------
// TDM: issue a 2-D tensor_load_to_lds. dim0 = contiguous (fastest) dimension,
// dim1 = rows, stride0 = elements between rows (f32). Optional LDS padding of
// 1 DWORD after every (1<<(pad_code+1)) DWORDs (pad_code=4 -> every 32 DWORDs,
// i.e. row stride 33 floats for a 32-float row: kills LDS bank conflicts).
// D# layout per CDNA5 ISA. Issued once per wave
// (EXEC ignored); completion tracked with TENSORcnt.
// ---------------------------------------------------------------------------
__device__ __forceinline__ void tdm_load_2d(const void* gptr, uint32_t lds_off,
                                            uint32_t dim0, uint32_t dim1,
                                            uint64_t stride0, bool pad,
                                            uint32_t pad_code)
{
    uint64_t ga = (uint64_t)(uintptr_t)gptr;
    u32x4 g0;
    g0[0] = 1u;                                   // count=1 (valid), user mode
    g0[1] = lds_off;                              // LDS byte address
    g0[2] = (uint32_t)ga;                         // global addr [95:64]
    g0[3] = (uint32_t)((ga >> 32) & 0x01FFFFFFu)  // global addr [120:96]
          | (2u << 30);                           // type=2 ("image")
    uint32_t w0 = (2u << 16);                     // data_size = 4 bytes
    if (pad) w0 |= (1u << 20) | (pad_code << 22); // pad_enable, pad_amount=1 DWORD
    i32x8 g1;
    g1[0] = (int)w0;
    g1[1] = (int)((dim0 & 0xFFFFu) << 16);                         // tensor_dim0 lo
    g1[2] = (int)(((dim0 >> 16) & 0xFFFFu) | ((dim1 & 0xFFFFu) << 16)); // t_dim0 hi | t_dim1 lo
    g1[3] = (int)(((dim1 >> 16) & 0xFFFFu) | ((dim0 & 0xFFFFu) << 16)); // t_dim1 hi | tile_dim0
    g1[4] = (int)(dim1 & 0xFFFFu);                                 // tile_dim1 | tile_dim2=0
    g1[5] = (int)(uint32_t)(stride0 & 0xFFFFFFFFu);                // tensor_dim0_stride lo
    g1[6] = (int)(uint32_t)((stride0 >> 32) & 0xFFFFu);            // stride hi | dim1_stride=0
    g1[7] = 0;
    i32x4 gz = {0, 0, 0, 0};                      // groups 2/3: no 3D+/iterate
    asm volatile("tensor_load_to_lds %0, %1, %2, %3"
                 :
                 : "s"(g0), "s"(g1), "s"(gz), "s"(gz)
                 : "memory");
}

// ---------------------------------------------------------------------------
// Tiled f32 GEMM using V_WMMA_F32_16X16X4_F32 with TDM-staged, double-buffered
// LDS tiles. C[M x N] = A[M x K] @ B.
//   BT=false: B row-major, element (k,n) at B[k*bsk + n]   (bsn must be 1)
//   BT=true : B transposed, element (k,n) at B[n*bsn + k]  (x @ W^T GEMMs)
// Optional accum[m*ldc+n] += result (builds I+A+A^2+... for free), bias[n].
// Block = 256 threads = 8 waves; block tile 128x128; wave tile 64x32
// (4 x 2 grid of 16x16 f32 accumulators). Wave 0 drives the TDM pipeline:
// prefetch tile t+1, s_wait_tensorcnt 2 (tile t done), barrier, compute.
// ---------------------------------------------------------------------------
template <bool BT>
__global__ __launch_bounds__(256) void gemm_f32_wmma(
    const float* __restrict__ A, int lda,
    const float* __restrict__ B, int bsk, int bsn,
    float* __restrict__ C, int ldc,
    float* __restrict__ accum,
    const float* __restrict__ bias,
    int M, int N, int K)
{
    extern __shared__ float smem[];
    const uint32_t AF = TM * (TK + 1);                 // 4224 floats (padded rows)
    const uint32_t BF = BT ? TN * (TK + 1) : TK * TN;  // padded iff k-contiguous

    const uint32_t ldsBase = (uint32_t)(uintptr_t)smem;

    const int tid   = threadIdx.x;
    const int wave  = tid >> 5;
    const int lane  = tid & 31;
    const int lhalf = lane >> 4;   // 0: lanes 0-15, 1: lanes 16-31
    const int l15   = lane & 15;

    const int blockM = blockIdx.y * TM;
    const int blockN = blockIdx.x * TN;
    const int wm = (wave >> 2) * 64;     // 0 or 64
    const int wn = (wave & 3) * 32;      // 0,32,64,96

    v8f acc[4][2];
    for (int i = 0; i < 4; ++i)
        for (int j = 0; j < 2; ++j)
            for (int t = 0; t < 8; ++t) acc[i][j][t] = 0.0f;

    auto issueTile = [&](int buf, int k0) {
        // A tile: rows = M (TM of them), contiguous along K (TK) -> pad to 33.
        tdm_load_2d(A + (size_t)blockM * lda + k0,
                    ldsBase + (uint32_t)(buf ? AF : 0) * 4u,
                    TK, TM, (uint64_t)lda, true, 4);
        if (!BT) {
            // B tile: rows = K (TK), contiguous along N (TN) -> no pad.
            tdm_load_2d(B + (size_t)k0 * bsk + blockN,
                        ldsBase + (2 * AF + (buf ? BF : 0)) * 4u,
                        TN, TK, (uint64_t)bsk, false, 0);
        } else {
            // B^T tile: rows = N (TN), contiguous along K (TK) -> pad to 33.
            tdm_load_2d(B + (size_t)blockN * bsn + k0,
                        ldsBase + (2 * AF + (buf ? BF : 0)) * 4u,
                        TK, TN, (uint64_t)bsn, true, 4);
        }
    };

    const int nT = K / TK;
    if (wave == 0) issueTile(0, 0);

    for (int it = 0; it < nT; ++it) {
        __syncthreads();   // all waves done with the buffer we are about to refill
        if (wave == 0) {
            if (it + 1 < nT) {
                issueTile((it + 1) & 1, (it + 1) * TK);
                __builtin_amdgcn_s_wait_tensorcnt(2);  // current tile landed
            } else {
                __builtin_amdgcn_s_wait_tensorcnt(0);
            }
        }
        __syncthreads();   // current tile visible to every wave
        // Compute buffer pointers from the runtime index (do NOT build pointer
        // arrays over __shared__ -- clang turns those into static initializers
        // with addrspacecast that ld.lld rejects).
        const int cur = it & 1;
        const float* sA = smem + (uint32_t)cur * AF;
        const float* sB = smem + 2 * AF + (uint32_t)cur * BF;

        #pragma unroll
        for (int kk = 0; kk < TK; kk += 4) {
            const int ka = kk + 2 * lhalf;  // lanes 0-15: K={ka,ka+1}; 16-31: +2
            v2f af[4];
            #pragma unroll
            for (int i = 0; i < 4; ++i) {
                int row = wm + i * 16 + l15;
                af[i][0] = sA[row * (TK + 1) + ka];
                af[i][1] = sA[row * (TK + 1) + ka + 1];
            }
            v2f bf[2];
            #pragma unroll
            for (int j = 0; j < 2; ++j) {
                int col = wn + j * 16 + l15;
                if (!BT) {
                    bf[j][0] = sB[ka * TN + col];
                    bf[j][1] = sB[(ka + 1) * TN + col];
                } else {
                    bf[j][0] = sB[col * (TK + 1) + ka];
                    bf[j][1] = sB[col * (TK + 1) + ka + 1];
                }
            }
            #pragma unroll
            for (int i = 0; i < 4; ++i)
                #pragma unroll
                for (int j = 0; j < 2; ++j)
                    acc[i][j] = __builtin_amdgcn_wmma_f32_16x16x4_f32(
                        false, af[i], false, bf[j], (short)0, acc[i][j],
                        false, false);
        }
    }

    // Epilogue. C/D layout: VGPR v -> M = v + 8*lhalf, N = l15.
    for (int i = 0; i < 4; ++i) {
        for (int j = 0; j < 2; ++j) {
            int n = blockN + wn + j * 16 + l15;
            #pragma unroll
            for (int v = 0; v < 8; ++v) {
                int m = blockM + wm + i * 16 + v + 8 * lhalf;
                float val = acc[i][j][v];
                if (bias) val += bias[n];
                C[(size_t)m * ldc + n] = val;
                if (accum) accum[(size_t)m * ldc + n] += val;
            }
        }
    }
}

// ---------------------------------------------------------------------------
// Elementwise / graph kernels
// ---------------------------------------------------------------------------
__global__ void zero_f32(float* p, size_t n) {
    size_t i = (size_t)blockIdx.x * blockDim.x + threadIdx.x;
    size_t stride = (size_t)gridDim.x * blockDim.x;
    for (; i < n; i += stride) p[i] = 0.0f;
}

__global__ void build_adj(const int* __restrict__ ei, float* __restrict__ adj) {
    int e = blockIdx.x * 256 + threadIdx.x;
    if (e >= kE) return;
    int s = ei[e], d = ei[kE + e];
    atomicAdd(&adj[(size_t)s * kN + d], 1.0f);
}

__global__ void add_eye(float* adj) {
    int i = blockIdx.x * 256 + threadIdx.x;
    if (i < kN) adj[(size_t)i * kN + i] += 1.0f;
}

// P1 = A (first walk step is I@A), result = I + A
__global__ void init_walk(const float* __restrict__ adj, float* __restrict__ P,
                          float* __restrict__ result) {
    size_t idx = (size_t)blockIdx.x * 256 + threadIdx.x;
    float a = adj[idx];
    P[idx] = a;
    int r = (int)(idx >> 11), c = (int)(idx & 2047);
    result[idx] = a + (r == c ? 1.0f : 0.0f);
}

// L2-normalize each row of the 2048x2048 result matrix (in place).
__global__ __launch_bounds__(256) void row_normalize(float* __restrict__ mat) {
    __shared__ float sm[256];
    int row = blockIdx.x, t = threadIdx.x;
    float* r = mat + (size_t)row * kN;
    float s = 0.0f;
    for (int i = t; i < kN; i += 256) { float v = r[i]; s += v * v; }
    sm[t] = s;
    __syncthreads();
    for (int o = 128; o > 0; o >>= 1) { if (t < o) sm[t] += sm[t + o]; __syncthreads(); }
    float norm = sqrtf(sm[0]);
    float inv = 1.0f / fmaxf(norm, 1e-12f);
    __syncthreads();
    for (int i = t; i < kN; i += 256) r[i] *= inv;
}

// s_i[n,h] = <xh[n,h,:], att[h,0:128]>, s_j with att[h,128:256]. 1 wave per (n,h).
__global__ __launch_bounds__(256) void compute_scores(
    const float* __restrict__ xh, const float* __restrict__ att,
    float* __restrict__ si, float* __restrict__ sj) {
    int w = blockIdx.x * 8 + (threadIdx.x >> 5);  // 0 .. N*H-1
    int lane = threadIdx.x & 31;
    int n = w >> 3, h = w & 7;
    const float4 xv = *(const float4*)(xh + (size_t)n * (kH * kC) + h * kC + lane * 4);
    const float4 ai = *(const float4*)(att + h * (2 * kC) + lane * 4);
    const float4 aj = *(const float4*)(att + h * (2 * kC) + kC + lane * 4);
    float pi = xv.x * ai.x + xv.y * ai.y + xv.z * ai.z + xv.w * ai.w;
    float pj = xv.x * aj.x + xv.y * aj.y + xv.z * aj.z + xv.w * aj.w;
    for (int off = 16; off >= 1; off >>= 1) {
        pi += __shfl_xor(pi, off, 32);
        pj += __shfl_xor(pj, off, 32);
    }
    if (lane == 0) { si[n * kH + h] = pi; sj[n * kH + h] = pj; }
}

// alpha[h,e] = leaky(s_i[dst,h] + s_j[src,h]) * mask[src,dst]
__global__ void compute_alpha(const int* __restrict__ ei, const float* __restrict__ si,
                              const float* __restrict__ sj, const float* __restrict__ mask,
                              float* __restrict__ alpha) {
    int e = blockIdx.x * 256 + threadIdx.x;
    if (e >= kE) return;
    int s = ei[e], d = ei[kE + e];
    float m = mask[(size_t)s * kN + d];
    #pragma unroll
    for (int h = 0; h < kH; ++h) {
        float a = si[d * kH + h] + sj[s * kH + h];
        a = (a >= 0.0f) ? a : NEG_SLOPE * a;
        alpha[(size_t)h * kE + e] = a * m;
    }
}

// Global softmax over all E edges, per head: two-pass (max, then exp-sum).
__global__ __launch_bounds__(256) void partial_max(const float* __restrict__ alpha,
                                                   float* __restrict__ pmax) {
    __shared__ float sm[256];
    int h = blockIdx.y, b = blockIdx.x, t = threadIdx.x;
    const float* a = alpha + (size_t)h * kE + (size_t)b * 1024;
    float m = -3.4e38f;
    #pragma unroll
    for (int k = 0; k < 4; ++k) m = fmaxf(m, a[t + 256 * k]);
    sm[t] = m;
    __syncthreads();
    for (int o = 128; o > 0; o >>= 1) { if (t < o) sm[t] = fmaxf(sm[t], sm[t + o]); __syncthreads(); }
    if (t == 0) pmax[h * 64 + b] = sm[0];
}

__global__ void reduce64_max(const float* __restrict__ pmax, float* __restrict__ hmax) {
    __shared__ float sm[64];
    int h = blockIdx.x, t = threadIdx.x;
    sm[t] = pmax[h * 64 + t];
    __syncthreads();
    for (int o = 32; o > 0; o >>= 1) { if (t < o) sm[t] = fmaxf(sm[t], sm[t + o]); __syncthreads(); }
    if (t == 0) hmax[h] = sm[0];
}

__global__ __launch_bounds__(256) void exp_psum(float* __restrict__ alpha,
                                                const float* __restrict__ hmax,
                                                float* __restrict__ psum) {
    __shared__ float sm[256];
    int h = blockIdx.y, b = blockIdx.x, t = threadIdx.x;
    float* a = alpha + (size_t)h * kE + (size_t)b * 1024;
    float mx = hmax[h];
    float s = 0.0f;
    #pragma unroll
    for (int k = 0; k < 4; ++k) {
        float ex = __expf(a[t + 256 * k] - mx);
        a[t + 256 * k] = ex;
        s += ex;
    }
    sm[t] = s;
    __syncthreads();
    for (int o = 128; o > 0; o >>= 1) { if (t < o) sm[t] += sm[t + o]; __syncthreads(); }
    if (t == 0) psum[h * 64 + b] = sm[0];
}

__global__ void reduce64_sum(const float* __restrict__ psum, float* __restrict__ hsum) {
    __shared__ float sm[64];
    int h = blockIdx.x, t = threadIdx.x;
    sm[t] = psum[h * 64 + t];
    __syncthreads();
    for (int o = 32; o > 0; o >>= 1) { if (t < o) sm[t] += sm[t + o]; __syncthreads(); }
    if (t == 0) hsum[h] = sm[0];
}

// segment_sum: out[dst,h,:] += xh[src,h,:] * softmax_weight. 1 wave per (e,h),
// 4 channels per lane (float4 load, 4 f32 atomics).
__global__ __launch_bounds__(256) void scatter_msg(
    const int* __restrict__ ei, const float* __restrict__ xh,
    const float* __restrict__ alpha, const float* __restrict__ hsum,
    float* __restrict__ out) {
    int gw = blockIdx.x * 8 + (threadIdx.x >> 5);
    int lane = threadIdx.x & 31;
    int e = gw >> 3, h = gw & 7;
    int s = ei[e], d = ei[kE + e];
    float w = alpha[(size_t)h * kE + e] / hsum[h];
    const float4 v = *(const float4*)(xh + (size_t)s * (kH * kC) + h * kC + lane * 4);
    float* o = out + (size_t)d * (kH * kC) + h * kC + lane * 4;
    atomicAdd(o + 0, v.x * w);
    atomicAdd(o + 1, v.y * w);
    atomicAdd(o + 2, v.z * w);
    atomicAdd(o + 3, v.w * w);
}

// dst[n,c] = relu(mean_h out[n,h,c]) (+ addend)
__global__ void finalize(const float* __restrict__ out, const float* __restrict__ addend,
                         float* __restrict__ dst) {
    int idx = blockIdx.x * 256 + threadIdx.x;  // N*C
    int n = idx >> 7, c = idx & 127;
    float s = 0.0f;
    #pragma unroll
    for (int h = 0; h < kH; ++h) s += out[(size_t)n * (kH * kC) + h * kC + c];
    s *= (1.0f / kH);
    s = s > 0.0f ? s : 0.0f;
    if (addend) s += addend[idx];
    dst[idx] = s;
}

// ---------------------------------------------------------------------------
extern "C" void kernel_launch(void* const* d_in, const int* in_sizes, int n_in,
                              void* d_out, int out_size, void* d_ws, size_t ws_size,
                              hipStream_t stream) {
    const float* x     = (const float*)d_in[0];
    const int*   ei    = (const int*)  d_in[1];
    const float* lw0   = (const float*)d_in[2];
    const float* att0  = (const float*)d_in[3];
    const float* lw1   = (const float*)d_in[4];
    const float* att1  = (const float*)d_in[5];
    const float* skw   = (const float*)d_in[6];
    const float* skb   = (const float*)d_in[7];
    float* outp = (float*)d_out;

    const size_t NN = (size_t)kN * kN;
    float* ws   = (float*)d_ws;
    float* adj  = ws;            // 16 MB
    float* bufA = ws + NN;       // 16 MB
    float* bufB = ws + 2 * NN;   // 16 MB
    float* mask = ws + 3 * NN;   // 16 MB (result -> normalized mask)

    // Reuse adj slab after the walk:
    float* xh     = adj;                       // N*H*C = 2097152 floats
    float* outacc = adj + (size_t)kN * kH * kC;
    // Reuse bufA slab for small buffers:
    float* si    = bufA;
    float* sj    = si + kN * kH;
    float* alpha = sj + kN * kH;
    float* pmax  = alpha + (size_t)kH * kE;
    float* psum  = pmax + 512;
    float* hmax  = psum + 512;
    float* hsum  = hmax + 8;
    float* hbuf  = hsum + 8;
    float* xskip = hbuf + (size_t)kN * kC;

    dim3 blk(256);
    const size_t AF = TM * (TK + 1);                       // floats
    const size_t shN  = (2 * AF + 2 * (size_t)(TK * TN)) * 4;        // BT=false
    const size_t shT  = (2 * AF + 2 * (size_t)(TN * (TK + 1))) * 4;  // BT=true
    auto gemmN = [&](const float* A, int lda, const float* B, int bsk,
                     float* C, int ldc, float* accum, const float* bias,
                     int M, int Nn, int K) {
        dim3 grid(Nn / TN, M / TM);
        gemm_f32_wmma<false><<<grid, blk, shN, stream>>>(A, lda, B, bsk, 1, C, ldc,
                                                         accum, bias, M, Nn, K);
    };
    auto gemmT = [&](const float* A, int lda, const float* B, int bsn,
                     float* C, int ldc, float* accum, const float* bias,
                     int M, int Nn, int K) {
        dim3 grid(Nn / TN, M / TM);
        gemm_f32_wmma<true><<<grid, blk, shT, stream>>>(A, lda, B, 1, bsn, C, ldc,
                                                        accum, bias, M, Nn, K);
    };

    // ---- 1. Walk mask: result = normalize_rows(I + A + A^2 + A^3 + A^4) ----
    zero_f32<<<2048, 256, 0, stream>>>(adj, NN);
    build_adj<<<kE / 256, 256, 0, stream>>>(ei, adj);
    add_eye<<<kN / 256, 256, 0, stream>>>(adj);
    init_walk<<<(int)(NN / 256), 256, 0, stream>>>(adj, bufA, mask);
    float* Pin = bufA;
    float* Pout = bufB;
    for (int it = 0; it < 3; ++it) {   // A^2, A^3, A^4, each accumulated into mask
        gemmN(Pin, kN, adj, kN, Pout, kN, mask, nullptr, kN, kN, kN);
        float* t = Pin; Pin = Pout; Pout = t;
    }
    row_normalize<<<kN, 256, 0, stream>>>(mask);

    // ---- 2. Skip connection: xskip = x @ skw^T + skb ----
    gemmT(x, kCIN, skw, kCIN, xskip, kC, nullptr, skb, kN, kC, kCIN);

    // ---- 3. Two GKAT layers ----
    auto layer = [&](const float* xin, const float* lw, const float* att,
                     const float* addend, float* dstp) {
        gemmT(xin, kCIN, lw, kCIN, xh, kH * kC, nullptr, nullptr, kN, kH * kC, kCIN);
        zero_f32<<<2048, 256, 0, stream>>>(outacc, (size_t)kN * kH * kC);
        compute_scores<<<kN * kH / 8, 256, 0, stream>>>(xh, att, si, sj);
        compute_alpha<<<kE / 256, 256, 0, stream>>>(ei, si, sj, mask, alpha);
        partial_max<<<dim3(64, kH), 256, 0, stream>>>(alpha, pmax);
        reduce64_max<<<kH, 64, 0, stream>>>(pmax, hmax);
        exp_psum<<<dim3(64, kH), 256, 0, stream>>>(alpha, hmax, psum);
        reduce64_sum<<<kH, 64, 0, stream>>>(psum, hsum);
        scatter_msg<<<kE * kH / 8, 256, 0, stream>>>(ei, xh, alpha, hsum, outacc);
        finalize<<<kN * kC / 256, 256, 0, stream>>>(outacc, addend, dstp);
    };
    layer(x,    lw0, att0, nullptr, hbuf);
    layer(hbuf, lw1, att1, xskip,   outp);

    (void)in_sizes; (void)n_in; (void)out_size; (void)ws_size;
}